// S5SSM_60361470378295
// MI455X (gfx1250) — compile-verified
//
#include <hip/hip_runtime.h>

// S5 SSM forward for MI455X (gfx1250, wave32).
// Pipeline: setup (discretize) -> WMMA GEMM (Bu = u @ B_bar^T, re+im planes)
//           -> 3-phase chunked linear recurrence scan (A constant per channel)
//           -> WMMA GEMM (ys = xs_re@Cre^T + xs_im@(-Cim)^T + D*u), x_last from phase 2.
// f32 WMMA (16x16x4): workload is HBM-bound (~380MB @ 23.3TB/s ~ 16us vs ~17 GFLOP),
// so full-fp32 matrix ops are free; no precision downcast.
// B/C operand panels are staged into LDS once per block via the Tensor Data Mover
// (tensor_load_to_lds + s_wait_tensorcnt), with TDM LDS-padding (row stride 260
// floats, 260 mod 64 = 4) so ds_load_b64 B-fragment reads are LDS bank-conflict-free.

#define LSEQ   32768
#define HDIM   256
#define PDIM   256
#define NCHUNK 256
#define CHUNK  128          // LSEQ / NCHUNK
#define WPB    4            // waves per block in GEMM kernels
#define LDSP   260          // padded LDS row stride (floats): 256 + 4

typedef __attribute__((ext_vector_type(2))) float v2f;
typedef __attribute__((ext_vector_type(8))) float v8f;
typedef __attribute__((ext_vector_type(4))) unsigned int v4u;
typedef __attribute__((ext_vector_type(8))) int v8i;
typedef __attribute__((ext_vector_type(4))) int v4i;

#if defined(__has_builtin)
#if __has_builtin(__builtin_amdgcn_tensor_load_to_lds) && \
    __has_builtin(__builtin_amdgcn_s_wait_tensorcnt)
#define S5_USE_TDM 1
#endif
#endif

static __device__ __forceinline__ v8f wmma_f32(v2f a, v2f b, v8f c) {
    // 8-arg form: (neg_a, A, neg_b, B, c_mod, C, reuse_a, reuse_b)
    return __builtin_amdgcn_wmma_f32_16x16x4_f32(false, a, false, b, (short)0, c, false, false);
}

#if defined(S5_USE_TDM)
// Issue one TDM descriptor: load a 16-row x 256-col f32 panel (row stride HDIM)
// from global into LDS at lds_addr, padding +4 DWORDs after every 256 DWORDs
// so LDS rows land at stride LDSP floats.
// This toolchain's builtin is the 6-arg form:
//   (uint32x4 g0, int32x8 g1, int32x4 g2, int32x4 g3, int32x8 extra, i32 cpol)
static __device__ __forceinline__ void tdm_load_panel(const float* gptr, unsigned lds_addr) {
    const unsigned long long ga = (unsigned long long)(uintptr_t)gptr;
    v4u g0;
    g0.x = 1u;                                   // count=1, user-mode descriptor
    g0.y = lds_addr;                             // lds_addr [63:32]
    g0.z = (unsigned)(ga & 0xFFFFFFFFu);         // global_addr [31:0]
    g0.w = (unsigned)((ga >> 32) & 0x01FFFFFFu)  // global_addr [56:32]
         | (2u << 30);                           // type = 2 ("image")
    v8i g1;
    g1[0] = (int)((2u << 16)                     // data_size = 4 bytes
                | (1u << 20)                     // pad_enable
                | (7u << 22)                     // pad_interval: 256 DWORDs
                | (3u << 25));                   // pad_amount: 4 DWORDs
    g1[1] = (int)((unsigned)(HDIM & 0xFFFF) << 16); // tensor_dim0[15:0] @ bits 63:48
    g1[2] = (int)(((HDIM >> 16) & 0xFFFF)        // tensor_dim0[31:16]
                | ((16u & 0xFFFF) << 16));       // tensor_dim1[15:0] = 16 rows
    g1[3] = (int)((unsigned)(HDIM & 0xFFFF) << 16); // tensor_dim1 hi=0; tile_dim0=256
    g1[4] = 16;                                  // tile_dim1 = 16 rows; tile_dim2 = 0
    g1[5] = HDIM;                                // tensor_dim0_stride[31:0] = 256
    g1[6] = 0;                                   // stride0 hi; tensor_dim1_stride lo
    g1[7] = 0;
    const v4i z4 = {0, 0, 0, 0};                 // groups 2/3 unused (2-D tile)
    const v8i z8 = {0, 0, 0, 0, 0, 0, 0, 0};
    __builtin_amdgcn_tensor_load_to_lds(g0, g1, z4, z4, z8, 0);
}
#endif

// Stage two 16x256 f32 panels into padded LDS (row stride LDSP), once per block.
static __device__ __forceinline__ void stage_panels(const float* __restrict__ g0,
                                                    const float* __restrict__ g1,
                                                    float* __restrict__ l0,
                                                    float* __restrict__ l1,
                                                    int nthreads) {
#if defined(S5_USE_TDM)
    if (threadIdx.x < 32) {                      // one wave issues both DMAs
        tdm_load_panel(g0, (unsigned)(uintptr_t)l0);
        tdm_load_panel(g1, (unsigned)(uintptr_t)l1);
        __builtin_amdgcn_s_wait_tensorcnt(0);
    }
    __syncthreads();
#else
    for (int i = threadIdx.x; i < 16 * HDIM; i += nthreads) {
        const int r = i >> 8;
        const int c = i & (HDIM - 1);
        l0[r * LDSP + c] = g0[r * HDIM + c];
        l1[r * LDSP + c] = g1[r * HDIM + c];
    }
    __syncthreads();
#endif
}

// ---------------------------------------------------------------------------
// Kernel 0: discretize. grid = P blocks, block = H threads.
// Computes Lambda_bar (per p), lam_bar^CHUNK (per p), B_bar planes (P,H),
// and deinterleaves C into Cre / -Cim planes (H,P).
// ---------------------------------------------------------------------------
__global__ void s5_setup(const float* __restrict__ Lambda,
                         const float* __restrict__ Bc,
                         const float* __restrict__ Cc,
                         const float* __restrict__ log_step,
                         float* __restrict__ lre,  float* __restrict__ lim,
                         float* __restrict__ mre,  float* __restrict__ mim,
                         float* __restrict__ Bre,  float* __restrict__ Bim,
                         float* __restrict__ Cre,  float* __restrict__ nCim) {
    const int p = blockIdx.x;
    const int h = threadIdx.x;

    const float lr = Lambda[2 * p];
    const float li = Lambda[2 * p + 1];
    const float step = __expf(log_step[p]);          // STEP_SCALE = 1
    const float s2 = 0.5f * step;

    // BL = 1 / (1 - s2*lam)
    const float dr = 1.0f - s2 * lr;
    const float di = -s2 * li;
    const float inv = 1.0f / (dr * dr + di * di);
    const float blr = dr * inv;
    const float bli = -di * inv;

    // Lambda_bar = BL * (1 + s2*lam)
    const float nr = 1.0f + s2 * lr;
    const float ni = s2 * li;
    const float lbr = blr * nr - bli * ni;
    const float lbi = blr * ni + bli * nr;

    if (h == 0) {
        lre[p] = lbr;
        lim[p] = lbi;
        // lam_bar^CHUNK via 7 complex squarings (2^7 = 128 = CHUNK)
        float wr = lbr, wi = lbi;
        #pragma unroll
        for (int i = 0; i < 7; ++i) {
            const float t = wr * wr - wi * wi;
            wi = 2.0f * wr * wi;
            wr = t;
        }
        mre[p] = wr;
        mim[p] = wi;
    }

    // B_bar[p,h] = (BL*step) * B[p,h]
    const float gr = blr * step;
    const float gi = bli * step;
    const float br = Bc[(p * HDIM + h) * 2];
    const float bi = Bc[(p * HDIM + h) * 2 + 1];
    Bre[p * HDIM + h] = gr * br - gi * bi;
    Bim[p * HDIM + h] = gr * bi + gi * br;

    // Deinterleave C (H,P,2): reuse (block=p, thread=h) as (col=p, row=h)
    Cre[h * PDIM + p]  =  Cc[(h * PDIM + p) * 2];
    nCim[h * PDIM + p] = -Cc[(h * PDIM + p) * 2 + 1];  // pre-negate: f32 WMMA has no B-negate
}

// ---------------------------------------------------------------------------
// GEMM 1: Bu_{re,im}(L,P) = u(L,H) @ B_bar_{re,im}(P,H)^T via f32 WMMA.
// B panels TDM-staged in LDS (shared across the 4 waves); one 16x16 tile/wave.
// ---------------------------------------------------------------------------
__global__ void s5_gemm_bu(const float* __restrict__ u,
                           const float* __restrict__ Bre,
                           const float* __restrict__ Bim,
                           float* __restrict__ bu_re,
                           float* __restrict__ bu_im) {
    __shared__ float lBre[16 * LDSP];
    __shared__ float lBim[16 * LDSP];

    const int wave = threadIdx.x >> 5;
    const int lane = threadIdx.x & 31;
    const int half = lane >> 4;
    const int mn   = lane & 15;
    const int kb   = half * 2;

    const int row0 = (blockIdx.x * WPB + wave) * 16;
    const int col0 = blockIdx.y * 16;

    stage_panels(Bre + col0 * HDIM, Bim + col0 * HDIM, lBre, lBim, WPB * 32);

    const float* ap  = u + (row0 + mn) * HDIM + kb;
    const float* bpr = &lBre[mn * LDSP + kb];
    const float* bpi = &lBim[mn * LDSP + kb];

    v8f acc_re = {};
    v8f acc_im = {};
    #pragma unroll 8
    for (int k = 0; k < HDIM; k += 4) {
        const v2f a  = *(const v2f*)(ap + k);
        const v2f br = *(const v2f*)(bpr + k);
        const v2f bi = *(const v2f*)(bpi + k);
        acc_re = wmma_f32(a, br, acc_re);
        acc_im = wmma_f32(a, bi, acc_im);
    }

    const int col = col0 + mn;
    #pragma unroll
    for (int v = 0; v < 8; ++v) {
        const int row = row0 + v + half * 8;   // ISA C/D layout
        bu_re[row * PDIM + col] = acc_re[v];
        bu_im[row * PDIM + col] = acc_im[v];
    }
}

// ---------------------------------------------------------------------------
// Scan phase 1: per-chunk local recurrence with zero carry-in -> chunk sums.
// grid = NCHUNK, block = P (channel per thread; loads coalesced across p).
// ---------------------------------------------------------------------------
__global__ void s5_scan_phase1(const float* __restrict__ bu_re,
                               const float* __restrict__ bu_im,
                               const float* __restrict__ lre,
                               const float* __restrict__ lim,
                               float* __restrict__ cs_re,
                               float* __restrict__ cs_im) {
    const int p = threadIdx.x;
    const int c = blockIdx.x;
    const float ar = lre[p], ai = lim[p];
    float xr = 0.0f, xi = 0.0f;
    const int base = c * CHUNK;
    #pragma unroll 4
    for (int i = 0; i < CHUNK; ++i) {
        const int idx = (base + i) * PDIM + p;
        const float br = bu_re[idx];
        const float bi = bu_im[idx];
        const float nr = ar * xr - ai * xi + br;
        const float ni = ar * xi + ai * xr + bi;
        xr = nr; xi = ni;
    }
    cs_re[c * PDIM + p] = xr;
    cs_im[c * PDIM + p] = xi;
}

// ---------------------------------------------------------------------------
// Scan phase 2: carry propagation across chunks (x' = s_c + lam^CHUNK * x),
// staged through LDS so the dependent chain runs at LDS latency.
// grid = P/32 blocks of 32 threads; 32 channels per block -> 64KB LDS.
// Also emits x_last (state after final chunk) straight into d_out tail.
// ---------------------------------------------------------------------------
__global__ void s5_scan_phase2(const float* __restrict__ cs_re,
                               const float* __restrict__ cs_im,
                               const float* __restrict__ mre,
                               const float* __restrict__ mim,
                               float* __restrict__ ci_re,
                               float* __restrict__ ci_im,
                               float* __restrict__ xlast) {
    __shared__ float sre[NCHUNK * 32];
    __shared__ float sim[NCHUNK * 32];
    const int lane = threadIdx.x;            // 0..31
    const int p = blockIdx.x * 32 + lane;

    for (int c = 0; c < NCHUNK; ++c) {
        sre[c * 32 + lane] = cs_re[c * PDIM + p];
        sim[c * 32 + lane] = cs_im[c * PDIM + p];
    }
    __syncthreads();

    const float wr = mre[p], wi = mim[p];
    float xr = 0.0f, xi = 0.0f;
    for (int c = 0; c < NCHUNK; ++c) {
        ci_re[c * PDIM + p] = xr;            // carry-in for chunk c (exclusive)
        ci_im[c * PDIM + p] = xi;
        const float sr = sre[c * 32 + lane];
        const float si = sim[c * 32 + lane];
        const float nr = sr + wr * xr - wi * xi;
        const float ni = si + wr * xi + wi * xr;
        xr = nr; xi = ni;
    }
    xlast[2 * p]     = xr;                   // x_last (P,2) real view
    xlast[2 * p + 1] = xi;
}

// ---------------------------------------------------------------------------
// Scan phase 3: replay each chunk with its carry-in; overwrite Bu -> xs in place.
// ---------------------------------------------------------------------------
__global__ void s5_scan_phase3(float* __restrict__ bu_re,
                               float* __restrict__ bu_im,
                               const float* __restrict__ lre,
                               const float* __restrict__ lim,
                               const float* __restrict__ ci_re,
                               const float* __restrict__ ci_im) {
    const int p = threadIdx.x;
    const int c = blockIdx.x;
    const float ar = lre[p], ai = lim[p];
    float xr = ci_re[c * PDIM + p];
    float xi = ci_im[c * PDIM + p];
    const int base = c * CHUNK;
    #pragma unroll 4
    for (int i = 0; i < CHUNK; ++i) {
        const int idx = (base + i) * PDIM + p;
        const float br = bu_re[idx];
        const float bi = bu_im[idx];
        const float nr = ar * xr - ai * xi + br;
        const float ni = ar * xi + ai * xr + bi;
        xr = nr; xi = ni;
        bu_re[idx] = xr;
        bu_im[idx] = xi;
    }
}

// ---------------------------------------------------------------------------
// GEMM 2: ys(L,H) = xs_re @ Cre^T + xs_im @ (-Cim)^T + D*u. Single acc chain.
// C panels TDM-staged in LDS like GEMM 1.
// ---------------------------------------------------------------------------
__global__ void s5_gemm_ys(const float* __restrict__ xs_re,
                           const float* __restrict__ xs_im,
                           const float* __restrict__ Cre,
                           const float* __restrict__ nCim,
                           const float* __restrict__ u,
                           const float* __restrict__ D,
                           float* __restrict__ ys) {
    __shared__ float lCre[16 * LDSP];
    __shared__ float lCim[16 * LDSP];

    const int wave = threadIdx.x >> 5;
    const int lane = threadIdx.x & 31;
    const int half = lane >> 4;
    const int mn   = lane & 15;
    const int kb   = half * 2;

    const int row0 = (blockIdx.x * WPB + wave) * 16;
    const int col0 = blockIdx.y * 16;

    stage_panels(Cre + col0 * PDIM, nCim + col0 * PDIM, lCre, lCim, WPB * 32);

    const float* apr = xs_re + (row0 + mn) * PDIM + kb;
    const float* api = xs_im + (row0 + mn) * PDIM + kb;
    const float* bpr = &lCre[mn * LDSP + kb];
    const float* bpi = &lCim[mn * LDSP + kb];

    v8f acc = {};
    #pragma unroll 8
    for (int k = 0; k < PDIM; k += 4) {
        const v2f ar = *(const v2f*)(apr + k);
        const v2f ai = *(const v2f*)(api + k);
        const v2f br = *(const v2f*)(bpr + k);
        const v2f bi = *(const v2f*)(bpi + k);
        acc = wmma_f32(ar, br, acc);
        acc = wmma_f32(ai, bi, acc);
    }

    const int col = col0 + mn;
    const float d = D[col];
    #pragma unroll
    for (int v = 0; v < 8; ++v) {
        const int row = row0 + v + half * 8;
        ys[row * HDIM + col] = acc[v] + d * u[row * HDIM + col];
    }
}

// ---------------------------------------------------------------------------
// Host-side launch. Inputs: u, prev_state(unused by the math: A[0]'s scaling
// never reaches xs/ys in the reference scan), Lambda, B, C, D, log_step.
// d_out = [ys (L*H) | x_last (P*2)], float32.
// ---------------------------------------------------------------------------
extern "C" void kernel_launch(void* const* d_in, const int* in_sizes, int n_in,
                              void* d_out, int out_size, void* d_ws, size_t ws_size,
                              hipStream_t stream) {
    const float* u        = (const float*)d_in[0];
    const float* Lambda   = (const float*)d_in[2];
    const float* Bc       = (const float*)d_in[3];
    const float* Cc       = (const float*)d_in[4];
    const float* D        = (const float*)d_in[5];
    const float* log_step = (const float*)d_in[6];

    float* out   = (float*)d_out;
    float* ys    = out;
    float* xlast = out + (size_t)LSEQ * HDIM;

    // Workspace layout (floats)
    float* ws = (float*)d_ws;
    size_t o = 0;
    float* lre   = ws + o; o += PDIM;
    float* lim   = ws + o; o += PDIM;
    float* mre   = ws + o; o += PDIM;
    float* mim   = ws + o; o += PDIM;
    float* Bre   = ws + o; o += (size_t)PDIM * HDIM;
    float* Bim   = ws + o; o += (size_t)PDIM * HDIM;
    float* Crep  = ws + o; o += (size_t)HDIM * PDIM;
    float* nCim  = ws + o; o += (size_t)HDIM * PDIM;
    float* cs_re = ws + o; o += (size_t)NCHUNK * PDIM;
    float* cs_im = ws + o; o += (size_t)NCHUNK * PDIM;
    float* ci_re = ws + o; o += (size_t)NCHUNK * PDIM;
    float* ci_im = ws + o; o += (size_t)NCHUNK * PDIM;
    float* bu_re = ws + o; o += (size_t)LSEQ * PDIM;
    float* bu_im = ws + o; o += (size_t)LSEQ * PDIM;   // total ~66 MB

    s5_setup<<<dim3(PDIM), dim3(HDIM), 0, stream>>>(
        Lambda, Bc, Cc, log_step, lre, lim, mre, mim, Bre, Bim, Crep, nCim);

    s5_gemm_bu<<<dim3(LSEQ / 16 / WPB, PDIM / 16), WPB * 32, 0, stream>>>(
        u, Bre, Bim, bu_re, bu_im);

    s5_scan_phase1<<<dim3(NCHUNK), dim3(PDIM), 0, stream>>>(
        bu_re, bu_im, lre, lim, cs_re, cs_im);

    s5_scan_phase2<<<dim3(PDIM / 32), dim3(32), 0, stream>>>(
        cs_re, cs_im, mre, mim, ci_re, ci_im, xlast);

    s5_scan_phase3<<<dim3(NCHUNK), dim3(PDIM), 0, stream>>>(
        bu_re, bu_im, lre, lim, ci_re, ci_im);

    s5_gemm_ys<<<dim3(LSEQ / 16 / WPB, HDIM / 16), WPB * 32, 0, stream>>>(
        bu_re, bu_im, Crep, nCim, u, D, ys);
}